// GCN_30116310680051
// MI455X (gfx1250) — compile-verified
//
#include <hip/hip_runtime.h>
#include <math.h>
#include <stdint.h>

typedef __attribute__((ext_vector_type(2))) float v2f;
typedef __attribute__((ext_vector_type(8))) float v8f;

#define NN 100000   // nodes
#define NE 3200000  // edges
#define NF 500      // input features
#define NH 16       // hidden
#define NC 40       // classes
#define MT ((NN + 15) / 16)  // 6250 M-tiles

// Hardware FP32 atomic add (global_atomic_add_f32, no CAS loop).
__device__ __forceinline__ void atomAddF(float* p, float v) {
    __hip_atomic_fetch_add(p, v, __ATOMIC_RELAXED, __HIP_MEMORY_SCOPE_AGENT);
}

// Async global->LDS copy of 16 bytes (gfx1250 ASYNCcnt path).
__device__ __forceinline__ void asyncCopyB128(void* lds_dst, const void* gsrc) {
    unsigned lds_off = (unsigned)(uintptr_t)lds_dst;         // low 32 bits = LDS byte addr
    unsigned long long ga = (unsigned long long)(uintptr_t)gsrc;
    asm volatile("global_load_async_to_lds_b128 %0, %1, off"
                 :: "v"(lds_off), "v"(ga) : "memory");
}
__device__ __forceinline__ void waitAsync0() {
    asm volatile("s_wait_asynccnt 0x0" ::: "memory");
}

// ---------------- degree / normalization ----------------
__global__ void k_deg_init(float* deg) {
    int i = blockIdx.x * blockDim.x + threadIdx.x;
    if (i < NN) deg[i] = 1.0f;  // self-loop
}

__global__ void k_deg_edges(const long long* __restrict__ ei, float* deg) {
    int e = blockIdx.x * blockDim.x + threadIdx.x;
    if (e < NE) {
        long long d = ei[NE + e];  // dst row
        atomAddF(&deg[d], 1.0f);
    }
}

__global__ void k_dinv(float* deg) {
    int i = blockIdx.x * blockDim.x + threadIdx.x;
    if (i < NN) deg[i] = rsqrtf(deg[i]);  // deg >= 1 always
}

// ---------------- GEMM1: H1 = X @ W1  (f32 WMMA 16x16x4) ----------------
// W1 (500x16 = 32 KB) staged once per block into LDS via async copies;
// B-operand then comes from the DS pipe while A streams from global.
__global__ void __launch_bounds__(128) k_gemm1(const float* __restrict__ X,
                                               const float* __restrict__ W1,
                                               float* __restrict__ H1) {
    __shared__ float sW1[NF * NH];  // 32000 B of the 320 KB WGP LDS

    // stage W1: 2000 x b128 async transfers across 128 threads
    for (int i = threadIdx.x; i < (NF * NH) / 4; i += 128)
        asyncCopyB128(&sW1[i * 4], W1 + (size_t)i * 4);
    waitAsync0();          // this wave's async ops complete
    __syncthreads();       // all waves' staging visible

    int lane  = threadIdx.x & 31;
    int wv    = threadIdx.x >> 5;
    int mtile = blockIdx.x * 4 + wv;
    if (mtile >= MT) return;                  // wave-uniform: EXEC stays all-1s
    int m0 = mtile * 16;
    int r  = lane & 15;                       // A row index / B col index
    int kh = (lane >> 4) << 1;                // 0 (lanes 0-15) or 2 (lanes 16-31)

    const float* xr = X + (size_t)(m0 + r) * NF + kh;
    const float* bw = sW1 + kh * NH + r;      // B: rows kh, kh+1 at col r
    v8f acc = {};
    #pragma unroll 5
    for (int k = 0; k < NF; k += 4) {
        v2f a, b;
        a.x = xr[k];                          // A: K = k+kh
        a.y = xr[k + 1];                      // A: K = k+kh+1
        b.x = bw[k * NH];                     // B: row K = k+kh,   col r (LDS)
        b.y = bw[k * NH + NH];                // B: row K = k+kh+1, col r (LDS)
        acc = __builtin_amdgcn_wmma_f32_16x16x4_f32(false, a, false, b,
                                                    (short)0, acc, false, false);
    }
    // C/D layout: VGPR j -> M = (lane>=16 ? 8 : 0)+j, N = lane&15
    int mr = (lane >> 4) << 3;
    float* out = H1 + (size_t)(m0 + mr) * NH + r;
    #pragma unroll
    for (int j = 0; j < 8; ++j) out[(size_t)j * NH] = acc[j];
}

// ---------------- aggregation 1 (into pre-activation G1) ----------------
__global__ void k_agg1_init(const float* __restrict__ H1,
                            const float* __restrict__ dinv,
                            float* __restrict__ G1) {
    int i = blockIdx.x * blockDim.x + threadIdx.x;
    if (i < NN * NH) {
        int node = i >> 4;
        float di = dinv[node];
        G1[i] = H1[i] * di * di;  // self-loop term
    }
}

__global__ void k_agg1_edges(const long long* __restrict__ ei,
                             const float* __restrict__ H1,
                             const float* __restrict__ dinv,
                             float* __restrict__ G1) {
    int t = blockIdx.x * blockDim.x + threadIdx.x;
    int e = t >> 4, f = t & 15;
    if (e < NE) {
        if (f == 0) __builtin_prefetch(ei + e + 4096, 0, 1);  // global_prefetch_b8
        long long s = ei[e];
        long long d = ei[NE + e];
        float nrm = dinv[s] * dinv[d];
        atomAddF(&G1[d * NH + f], H1[s * NH + f] * nrm);
    }
}

// ---------------- GEMM2: H2 = relu(G1 + b1) @ W2 (f32 WMMA, fused bias/ReLU) --
__global__ void __launch_bounds__(128) k_gemm2(const float* __restrict__ G1,
                                               const float* __restrict__ b1,
                                               const float* __restrict__ W2,
                                               float* __restrict__ H2) {
    int lane = threadIdx.x & 31;
    int wv   = threadIdx.x >> 5;
    int t    = blockIdx.x * 4 + wv;
    if (t >= MT * 3) return;                  // wave-uniform exit
    int mtile = t / 3, ntile = t % 3;
    int m0 = mtile * 16;
    int r  = lane & 15;
    int kh = (lane >> 4) << 1;
    int n  = ntile * 16 + r;
    bool nok = (n < NC);

    const float* ar = G1 + (size_t)(m0 + r) * NH + kh;
    v8f acc = {};
    #pragma unroll
    for (int k = 0; k < NH; k += 4) {
        v2f a, b;
        a.x = fmaxf(ar[k]     + b1[k + kh],     0.0f);
        a.y = fmaxf(ar[k + 1] + b1[k + kh + 1], 0.0f);
        b.x = nok ? W2[(k + kh) * NC + n]     : 0.0f;   // zero-pad cols >= 40
        b.y = nok ? W2[(k + kh + 1) * NC + n] : 0.0f;
        acc = __builtin_amdgcn_wmma_f32_16x16x4_f32(false, a, false, b,
                                                    (short)0, acc, false, false);
    }
    if (nok) {
        int mr = (lane >> 4) << 3;
        float* out = H2 + (size_t)(m0 + mr) * NC + n;
        #pragma unroll
        for (int j = 0; j < 8; ++j) out[(size_t)j * NC] = acc[j];
    }
}

// ---------------- aggregation 2 (into d_out, bias pre-added) ----------------
__global__ void k_agg2_init(const float* __restrict__ H2,
                            const float* __restrict__ dinv,
                            const float* __restrict__ b2,
                            float* __restrict__ out) {
    int i = blockIdx.x * blockDim.x + threadIdx.x;
    if (i < NN * NC) {
        int node = i / NC;
        int c = i - node * NC;
        float di = dinv[node];
        out[i] = H2[i] * di * di + b2[c];
    }
}

__global__ void k_agg2_edges(const long long* __restrict__ ei,
                             const float* __restrict__ H2,
                             const float* __restrict__ dinv,
                             float* __restrict__ out) {
    int lane = threadIdx.x & 31;
    int e = (blockIdx.x * blockDim.x + threadIdx.x) >> 5;
    if (e >= NE) return;
    if (lane == 0) __builtin_prefetch(ei + e + 2048, 0, 1);
    long long s = ei[e];
    long long d = ei[NE + e];
    float nrm = dinv[s] * dinv[d];
    const float* hs = H2 + s * (long long)NC;
    float* od = out + d * (long long)NC;
    atomAddF(&od[lane], hs[lane] * nrm);
    if (lane < NC - 32) atomAddF(&od[lane + 32], hs[lane + 32] * nrm);
}

// ---------------- log_softmax over 40 classes, wave per node ----------------
__global__ void k_logsoftmax(float* __restrict__ out) {
    int lane = threadIdx.x & 31;
    int i = (blockIdx.x * blockDim.x + threadIdx.x) >> 5;
    if (i >= NN) return;
    float* row = out + (size_t)i * NC;
    float v0 = row[lane];
    float v1 = (lane < NC - 32) ? row[lane + 32] : -__builtin_inff();
    float m = fmaxf(v0, v1);
    #pragma unroll
    for (int off = 16; off > 0; off >>= 1) m = fmaxf(m, __shfl_xor(m, off, 32));
    float s = expf(v0 - m) + ((lane < NC - 32) ? expf(v1 - m) : 0.0f);
    #pragma unroll
    for (int off = 16; off > 0; off >>= 1) s += __shfl_xor(s, off, 32);
    float ls = logf(s);
    row[lane] = v0 - m - ls;
    if (lane < NC - 32) row[lane + 32] = v1 - m - ls;
}

extern "C" void kernel_launch(void* const* d_in, const int* in_sizes, int n_in,
                              void* d_out, int out_size, void* d_ws, size_t ws_size,
                              hipStream_t stream) {
    (void)in_sizes; (void)n_in; (void)out_size; (void)ws_size;
    const float*     X  = (const float*)d_in[0];
    const long long* EI = (const long long*)d_in[1];  // int64 [2, NE]
    const float*     W1 = (const float*)d_in[2];
    const float*     B1 = (const float*)d_in[3];
    const float*     W2 = (const float*)d_in[4];
    const float*     B2 = (const float*)d_in[5];
    float* OUT = (float*)d_out;

    float* ws   = (float*)d_ws;
    float* dinv = ws;                          // NN floats
    float* H1   = dinv + NN;                   // NN*NH
    float* G1   = H1 + (size_t)NN * NH;        // NN*NH
    float* H2   = G1 + (size_t)NN * NH;        // NN*NC

    // normalization
    k_deg_init <<<(NN + 255) / 256, 256, 0, stream>>>(dinv);
    k_deg_edges<<<(NE + 255) / 256, 256, 0, stream>>>(EI, dinv);
    k_dinv     <<<(NN + 255) / 256, 256, 0, stream>>>(dinv);
    // layer 1
    k_gemm1     <<<(MT + 3) / 4, 128, 0, stream>>>(X, W1, H1);
    k_agg1_init <<<(NN * NH + 255) / 256, 256, 0, stream>>>(H1, dinv, G1);
    k_agg1_edges<<<(NE * 16 + 255) / 256, 256, 0, stream>>>(EI, H1, dinv, G1);
    // layer 2 (bias+ReLU fused into GEMM2 A-operand)
    k_gemm2     <<<(MT * 3 + 3) / 4, 128, 0, stream>>>(G1, B1, W2, H2);
    k_agg2_init <<<(NN * NC + 255) / 256, 256, 0, stream>>>(H2, dinv, B2, OUT);
    k_agg2_edges<<<(NE * 32 + 255) / 256, 256, 0, stream>>>(EI, H2, dinv, OUT);
    // fused log_softmax
    k_logsoftmax<<<(NN * 32 + 255) / 256, 256, 0, stream>>>(OUT);
}